// DetectionLoss_52888227283608
// MI455X (gfx1250) — compile-verified
//
#include <hip/hip_runtime.h>
#include <math.h>

// Problem constants (from reference): B=16, C=81, H=W=128 -> A=16384, N=64
#define B_    16
#define C_    81
#define A_    16384
#define N_    64
#define TPB   256
#define TILES (A_ / TPB)     // 64 blocks per image
#define NBLK  (B_ * TILES)   // 1024 total blocks

#define LOG2E 1.44269504088896340736f
#define LN2   0.69314718055994530942f

typedef __attribute__((ext_vector_type(2))) float v2f;
typedef __attribute__((ext_vector_type(8))) float v8f;

// ---------------------------------------------------------------------------
// Exact f32 wave(32) sum via V_WMMA_F32_16X16X4_F32 with B = ones:
// A-layout (16x4 f32): lanes 0-15 -> {VGPR0:K0, VGPR1:K1}, lanes 16-31 -> {K2,K3}.
// With a[0]=v, a[1]=0: rowsum(m) = v(m) + v(m+16).  D VGPR r: lanes0-15 = row r,
// lanes16-31 = row r+8.  Sum the 8 D regs per lane, then fold with shfl_xor(16).
// ---------------------------------------------------------------------------
__device__ __forceinline__ float wave_sum_wmma(float v) {
  v2f a; a[0] = v;    a[1] = 0.0f;
  v2f b; b[0] = 1.0f; b[1] = 1.0f;
  v8f acc = {};
  acc = __builtin_amdgcn_wmma_f32_16x16x4_f32(
      /*neg_a=*/false, a, /*neg_b=*/false, b,
      /*c_mod=*/(short)0, acc, /*reuse_a=*/false, /*reuse_b=*/false);
  float t = acc[0] + acc[1] + acc[2] + acc[3] + acc[4] + acc[5] + acc[6] + acc[7];
  t += __shfl_xor(t, 16, 32);
  return t;  // full 32-lane sum, valid in every lane
}

// Exact sum of 64 consecutive floats using one WMMA (lane l supplies arr[l],
// arr[32+l]; the 16x4 A tile then covers all 64 elements exactly once).
__device__ __forceinline__ float reduce64_wmma(const float* __restrict__ arr, int lane) {
  v2f a; a[0] = arr[lane]; a[1] = arr[32 + lane];
  v2f b; b[0] = 1.0f;      b[1] = 1.0f;
  v8f acc = {};
  acc = __builtin_amdgcn_wmma_f32_16x16x4_f32(false, a, false, b, (short)0, acc, false, false);
  float t = acc[0] + acc[1] + acc[2] + acc[3] + acc[4] + acc[5] + acc[6] + acc[7];
  t += __shfl_xor(t, 16, 32);
  return t;
}

// Block (256 = 8 waves) reduction -> thread 0 stores to *dst. Deterministic:
// in-wave order fixed by WMMA, cross-wave fold is a fixed 8-term serial sum.
__device__ __forceinline__ void block_reduce_store(float v, float* sred,
                                                   float* __restrict__ dst) {
  float w = wave_sum_wmma(v);
  const int lane = threadIdx.x & 31;
  const int wid  = threadIdx.x >> 5;
  __syncthreads();               // protect sred reuse across successive calls
  if (lane == 0) sred[wid] = w;
  __syncthreads();
  if (threadIdx.x == 0) {
    float r = 0.0f;
#pragma unroll
    for (int i = 0; i < 8; ++i) r += sred[i];
    *dst = r;
  }
}

__device__ __forceinline__ float smooth_l1(float d) {
  float ad = fabsf(d);
  return (ad < 1.0f) ? (0.5f * d * d) : (ad - 0.5f);
}

// ---------------------------------------------------------------------------
// Fused kernel: one block = 256 anchors of one image.
//  1) IoU/argmax vs 64 GT boxes (boxes as float4 + precomputed areas in LDS;
//     division replaced by v_rcp_f32 * mul)
//  2) smooth-L1 regression term vs assigned box
//  3) register-resident 81-class log-softmax; exp/log on the TRANS pipe
//  4) four WMMA-based block reductions -> per-block partials in d_ws
// ---------------------------------------------------------------------------
__global__ __launch_bounds__(TPB) void det_loss_main(
    const float* __restrict__ bbox_pred,   // (B,4,H,W)
    const float* __restrict__ class_pred,  // (B,C,H,W)
    const float* __restrict__ boxes,       // (B,N,4)
    const int*   __restrict__ labels,      // (B,N)
    float* __restrict__ npos_part, float* __restrict__ reg_part,
    float* __restrict__ cls_part,  float* __restrict__ zero_part) {
  __shared__ float4 sbox[N_];    // GT boxes, one b128 broadcast read per j
  __shared__ float  sga[N_];     // GT areas, hoisted out of the j-loop
  __shared__ int    slab[N_];
  __shared__ float  sred[8];

  const int blk  = blockIdx.x;
  const int b    = blk >> 6;       // / TILES
  const int tile = blk & (TILES - 1);
  const int tid  = threadIdx.x;
  const int p    = tile * TPB + tid;

  if (tid < N_) {
    float4 g = ((const float4*)boxes)[b * N_ + tid];
    sbox[tid] = g;
    sga[tid]  = (g.z - g.x) * (g.w - g.y);
    slab[tid] = labels[b * N_ + tid];
  }
  __syncthreads();

  // --- predicted box (channel-major layout, coalesced, read-once -> NT) ---
  const float* bb = bbox_pred + (size_t)b * 4 * A_ + p;
  const float px1 = __builtin_nontemporal_load(bb + 0 * A_);
  const float py1 = __builtin_nontemporal_load(bb + 1 * A_);
  const float px2 = __builtin_nontemporal_load(bb + 2 * A_);
  const float py2 = __builtin_nontemporal_load(bb + 3 * A_);
  const float pa  = (px2 - px1) * (py2 - py1);

  // --- IoU argmax over 64 GT boxes (first-max wins, matches jnp.argmax) ---
  float best = -__builtin_inff();
  int bestj = 0;
#pragma unroll 4
  for (int j = 0; j < N_; ++j) {
    const float4 g = sbox[j];
    const float iw = fminf(px2, g.z) - fmaxf(px1, g.x);
    const float ih = fminf(py2, g.w) - fmaxf(py1, g.y);
    const float inter = fmaxf(iw, 0.0f) * fmaxf(ih, 0.0f);
    const float uni = (pa + sga[j]) - inter;
    // IEEE-div -> rcp*mul: sign/inf preserved, 0*inf = NaN matches 0/0 = NaN
    const float iou = inter * __builtin_amdgcn_rcpf(uni);
    if (iou > best) { best = iou; bestj = j; }
  }
  const bool  pos  = best > 0.5f;
  const float posf = pos ? 1.0f : 0.0f;

  // --- smooth-L1 vs assigned box (masked by pos) ---
  const float4 ab = sbox[bestj];
  const float regv = posf * (smooth_l1(px1 - ab.x) + smooth_l1(py1 - ab.y) +
                             smooth_l1(px2 - ab.z) + smooth_l1(py2 - ab.w));
  const int lbl = pos ? slab[bestj] : 0;

  // --- 81-class log-softmax, values held in VGPRs (two-pass, f32) ---
  const float* cp = class_pred + (size_t)b * C_ * A_ + p;
  float xs[C_];
  float m = -__builtin_inff();
#pragma unroll
  for (int c = 0; c < C_; ++c) {
    const float x = __builtin_nontemporal_load(cp + (size_t)c * A_);
    xs[c] = x;
    m = fmaxf(m, x);
  }
  // x[label] via one extra (L2-hot) load instead of 81 cmp+cndmask selects
  const float xl = __builtin_nontemporal_load(cp + (size_t)lbl * A_);

  const float nm2 = -m * LOG2E;          // exp(x-m) = exp2(x*log2e - m*log2e)
  float s = 0.0f;
#pragma unroll
  for (int c = 0; c < C_; ++c)
    s += __builtin_amdgcn_exp2f(fmaf(xs[c], LOG2E, nm2));
  const float lse = fmaf(__builtin_amdgcn_logf(s), LN2, m);  // m + ln(s)
  const float cev = pos ? (lse - xl) : 0.0f;  // -log_softmax[label] * posf
  const float zv  = lse - xs[0];              // -log_softmax[0]

  // --- per-block partials via WMMA reductions ---
  block_reduce_store(posf, sred, npos_part + blk);
  block_reduce_store(regv, sred, reg_part + blk);
  block_reduce_store(cev,  sred, cls_part + blk);
  block_reduce_store(zv,   sred, zero_part + blk);
}

// ---------------------------------------------------------------------------
// Final fold: one wave, 16 images x 4 quantities, each 64 partials -> one WMMA.
// ---------------------------------------------------------------------------
__global__ __launch_bounds__(32) void det_loss_final(
    const float* __restrict__ npos_part, const float* __restrict__ reg_part,
    const float* __restrict__ cls_part,  const float* __restrict__ zero_part,
    float* __restrict__ out) {
  const int lane = threadIdx.x;
  float acc = 0.0f;
  for (int b = 0; b < B_; ++b) {
    const float np = reduce64_wmma(npos_part + b * TILES, lane);
    const float rg = reduce64_wmma(reg_part  + b * TILES, lane);
    const float cl = reduce64_wmma(cls_part  + b * TILES, lane);
    const float zr = reduce64_wmma(zero_part + b * TILES, lane);
    const float denom = fmaxf(np, 1.0f);
    const float per = (np > 0.0f) ? (cl / denom + rg / (denom * 4.0f))
                                  : (zr * (1.0f / (float)A_));
    acc += per;
  }
  if (lane == 0) out[0] = acc * (1.0f / (float)B_);
}

extern "C" void kernel_launch(void* const* d_in, const int* in_sizes, int n_in,
                              void* d_out, int out_size, void* d_ws, size_t ws_size,
                              hipStream_t stream) {
  (void)in_sizes; (void)n_in; (void)out_size; (void)ws_size;
  const float* bbox_pred  = (const float*)d_in[0];
  const float* class_pred = (const float*)d_in[1];
  const float* boxes      = (const float*)d_in[2];
  const int*   labels     = (const int*)d_in[3];

  float* ws        = (float*)d_ws;          // 4 * NBLK floats = 16 KB scratch
  float* npos_part = ws;
  float* reg_part  = ws + NBLK;
  float* cls_part  = ws + 2 * NBLK;
  float* zero_part = ws + 3 * NBLK;

  det_loss_main<<<NBLK, TPB, 0, stream>>>(bbox_pred, class_pred, boxes, labels,
                                          npos_part, reg_part, cls_part, zero_part);
  det_loss_final<<<1, 32, 0, stream>>>(npos_part, reg_part, cls_part, zero_part,
                                       (float*)d_out);
}